// LightGCN_8598524527001
// MI455X (gfx1250) — compile-verified
//
#include <hip/hip_runtime.h>
#include <hip/hip_bf16.h>

// LightGCN propagation for MI455X (gfx1250, wave32).
//   1) histogram out-degree (normalization) + in-degree (CSR) with global u32 atomics
//   2) dis[i] = rsqrt(outdeg[i] + 1)
//   3) single-WG tiled scan of in-degree -> rowptr
//   4) fill CSR-by-target with interleaved records {src, norm} (one b64 per edge)
//   5) 3x pull layers: one wave32 per node, float2 per lane (32*2 = 64 dims).
//      Edge batches are staged to LDS with gfx1250 async global->LDS DMA
//      (global_load_async_to_lds_b64 + s_wait_asynccnt). Inner loop: broadcast
//      ds_load_b64. acc init (layer 1) and final *0.25 (layer 3) fused in.

#define EMB_D 64

#if defined(__has_builtin)
#  if __has_builtin(__builtin_amdgcn_global_load_async_to_lds_b64)
#    define HAVE_ASYNC_LDS 1
#  endif
#endif
#ifndef HAVE_ASYNC_LDS
#  define HAVE_ASYNC_LDS 0
#endif

// Native clang vector type + address-space-qualified pointers for the builtin:
// param0 is 'int __vector(2) __device__ *' (global), param1 is the LDS side.
typedef int v2i_nat __attribute__((ext_vector_type(2)));
typedef __attribute__((address_space(1))) v2i_nat* gptr_v2i;
typedef __attribute__((address_space(3))) v2i_nat* lptr_v2i;

__device__ __forceinline__ void wait_asynccnt0() {
#if defined(__has_builtin) && __has_builtin(__builtin_amdgcn_s_wait_asynccnt)
    __builtin_amdgcn_s_wait_asynccnt(0);
#else
    asm volatile("s_wait_asynccnt 0x0" ::: "memory");
#endif
}

// ---------------------------------------------------------------------------
// Pull kernel (placed first so the disasm snippet shows it).
// One wave32 per node; lane owns dims [2*lane, 2*lane+1] as float2.
// firstLayer: acc = (emb_row + x_new) * accScale   (emb row already in regs)
// else:       acc = (acc    + x_new) * accScale
// ---------------------------------------------------------------------------
__global__ __launch_bounds__(256) void lgcn_pull(const float* __restrict__ xin,
                                                 float* __restrict__ xout,
                                                 float* __restrict__ acc,
                                                 const int* __restrict__ rowptr,
                                                 const int2* __restrict__ rec,
                                                 const float* __restrict__ dis,
                                                 int N, int firstLayer, float accScale) {
    __shared__ int2 stage[8][32];               // 8 waves/block * 32 edges * 8B = 2 KB
    int gtid = blockIdx.x * blockDim.x + threadIdx.x;
    int node = gtid >> 5;
    int lane = gtid & 31;
    if (node >= N) return;
    int2* myStage = stage[threadIdx.x >> 5];

    const float2* xv = (const float2*)xin;
    float  d  = dis[node];
    float2 xs = xv[(size_t)node * 32 + lane];
    float2 av;                                   // self-loop term: dis[i]^2 * x[i]
    av.x = d * d * xs.x;
    av.y = d * d * xs.y;

    int beg = rowptr[node];
    int end = rowptr[node + 1];
    for (int t = beg; t < end; t += 32) {
        int idx = t + lane;
#if HAVE_ASYNC_LDS
        if (idx < end) {
            __builtin_amdgcn_global_load_async_to_lds_b64(
                (gptr_v2i)(rec + idx),           // global source (addrspace 1)
                (lptr_v2i)(myStage + lane),      // LDS destination (addrspace 3)
                0, 0);
        }
        wait_asynccnt0();
#else
        if (idx < end) myStage[lane] = rec[idx];
#endif
        if (idx < end) {
            // gfx1250 global_prefetch_b8: warm this lane's staged gather row
            int2 e0 = myStage[lane];
            __builtin_prefetch(&xv[(size_t)e0.x * 32 + lane], 0, 0);
        }
        int cnt = end - t;
        if (cnt > 32) cnt = 32;
        for (int k = 0; k < cnt; ++k) {
            int2  er = myStage[k];               // broadcast ds_load_b64 (all lanes same addr)
            float wk = __int_as_float(er.y);
            float2 g = xv[(size_t)er.x * 32 + lane];
            av.x = fmaf(wk, g.x, av.x);
            av.y = fmaf(wk, g.y, av.y);
        }
    }

    ((float2*)xout)[(size_t)node * 32 + lane] = av;

    float2* a2 = (float2*)acc;
    float2 base = firstLayer ? xs : a2[(size_t)node * 32 + lane];
    float2 o;
    o.x = (base.x + av.x) * accScale;
    o.y = (base.y + av.y) * accScale;
    a2[(size_t)node * 32 + lane] = o;
}

// ---------------------------------------------------------------------------
__global__ void lgcn_count(const int* __restrict__ row, const int* __restrict__ col,
                           int* __restrict__ outdeg, int* __restrict__ indeg, int E) {
    int e = blockIdx.x * blockDim.x + threadIdx.x;
    if (e < E) {
        atomicAdd(&outdeg[row[e]], 1);
        atomicAdd(&indeg[col[e]], 1);
    }
}

__global__ void lgcn_dis(const int* __restrict__ outdeg, float* __restrict__ dis, int N) {
    int i = blockIdx.x * blockDim.x + threadIdx.x;
    if (i < N) dis[i] = rsqrtf((float)(outdeg[i] + 1));  // deg >= 1 always (self-loop)
}

// Single-workgroup tiled Hillis-Steele scan: rowptr[0]=0, rowptr[i+1]=sum(counts[0..i]).
__global__ __launch_bounds__(1024) void lgcn_scan(const int* __restrict__ counts,
                                                  int* __restrict__ rowptr, int n) {
    __shared__ int sh[1024];
    __shared__ int s_carry;
    int tid = threadIdx.x;
    if (tid == 0) { s_carry = 0; rowptr[0] = 0; }
    __syncthreads();
    for (int base = 0; base < n; base += 1024) {
        int i = base + tid;
        sh[tid] = (i < n) ? counts[i] : 0;
        __syncthreads();
        for (int offs = 1; offs < 1024; offs <<= 1) {
            int t = (tid >= offs) ? sh[tid - offs] : 0;
            __syncthreads();
            sh[tid] += t;
            __syncthreads();
        }
        int incl = sh[tid] + s_carry;     // reads s_carry
        if (i < n) rowptr[i + 1] = incl;
        __syncthreads();                  // all reads of s_carry done before update
        if (tid == 1023) s_carry = incl;
        __syncthreads();
    }
}

__global__ void lgcn_fill(const int* __restrict__ row, const int* __restrict__ col,
                          const float* __restrict__ ew, const float* __restrict__ dis,
                          const int* __restrict__ rowptr, int* __restrict__ cursor,
                          int2* __restrict__ rec, int E) {
    int e = blockIdx.x * blockDim.x + threadIdx.x;
    if (e < E) {
        int r = row[e], c = col[e];
        int pos  = atomicAdd(&cursor[c], 1);
        int slot = rowptr[c] + pos;
        int2 rc;
        rc.x = r;
        rc.y = __float_as_int(dis[r] * ew[e] * dis[c]);
        rec[slot] = rc;                    // one aligned 8B store per edge
    }
}

// ---------------------------------------------------------------------------
extern "C" void kernel_launch(void* const* d_in, const int* in_sizes, int n_in,
                              void* d_out, int out_size, void* d_ws, size_t ws_size,
                              hipStream_t stream) {
    const float* emb = (const float*)d_in[0];   // [N, 64]
    const float* ew  = (const float*)d_in[1];   // [E]
    const int*   ei  = (const int*)d_in[2];     // [2, E] row-major

    const int N = in_sizes[0] / EMB_D;
    const int E = in_sizes[1];
    const int* row = ei;        // edge_index[0] = source
    const int* col = ei + E;    // edge_index[1] = target

    // Workspace layout (256B aligned slices): ~65.6 MB total for N=100K, E=1.6M
    char* ws = (char*)d_ws;
    size_t off = 0;
    auto aln = [](size_t x) { return (x + 255) & ~(size_t)255; };
    int*   outdeg = (int*)(ws + off);   off = aln(off + (size_t)N * 4);
    int*   indeg  = (int*)(ws + off);   off = aln(off + (size_t)N * 4);   // reused as cursor
    int*   rowptr = (int*)(ws + off);   off = aln(off + (size_t)(N + 1) * 4);
    float* dis    = (float*)(ws + off); off = aln(off + (size_t)N * 4);
    int2*  rec    = (int2*)(ws + off);  off = aln(off + (size_t)E * 8);
    float* xb1    = (float*)(ws + off); off = aln(off + (size_t)N * EMB_D * 4);
    float* xb2    = (float*)(ws + off); off = aln(off + (size_t)N * EMB_D * 4);

    float* acc = (float*)d_out;

    (void)hipMemsetAsync(outdeg, 0, (size_t)N * 4, stream);
    (void)hipMemsetAsync(indeg,  0, (size_t)N * 4, stream);

    int eb = (E + 255) / 256;
    int nb = (N + 255) / 256;

    lgcn_count<<<eb, 256, 0, stream>>>(row, col, outdeg, indeg, E);
    lgcn_dis<<<nb, 256, 0, stream>>>(outdeg, dis, N);
    lgcn_scan<<<1, 1024, 0, stream>>>(indeg, rowptr, N);
    (void)hipMemsetAsync(indeg, 0, (size_t)N * 4, stream);   // reuse as fill cursor
    lgcn_fill<<<eb, 256, 0, stream>>>(row, col, ew, dis, rowptr, indeg, rec, E);

    int pb = (N * 32 + 255) / 256;   // one wave32 per node
    lgcn_pull<<<pb, 256, 0, stream>>>(emb, xb1, acc, rowptr, rec, dis, N, 1, 1.0f);
    lgcn_pull<<<pb, 256, 0, stream>>>(xb1, xb2, acc, rowptr, rec, dis, N, 0, 1.0f);
    lgcn_pull<<<pb, 256, 0, stream>>>(xb2, xb1, acc, rowptr, rec, dis, N, 0, 0.25f);
}